// DecoderRNN_87892210745506
// MI455X (gfx1250) — compile-verified
//
#include <hip/hip_runtime.h>
#include <hip/hip_bf16.h>
#include <math.h>

typedef __attribute__((ext_vector_type(2))) float v2f;
typedef __attribute__((ext_vector_type(8))) float v8f;

#define H 1024
#define L 4096
#define VOCAB 50257

// ---- workspace layout (floats) ----
#define WS_X     0          // 1024   embedding row
#define WS_GI    1024       // 3072   w_ih@x + b_ih
#define WS_GH    4096       // 3072   w_hh@h + b_hh
#define WS_HN    7168       // 1024   h_new
#define WS_SC    8192       // 4096   scores -> attn (in place)
#define WS_CP    12288      // 16*1024 ctx partials
#define WS_CC    28672      // 2048   concat [ctx, h_new]
#define WS_LG    30720      // 50257  logits
#define WS_BMAX  81024      // 64
#define WS_BSUM  81088      // 64
#define WS_G     81152      // 2      (gmax, log(sumexp))

// =====================================================================
// Generic mat-vec via V_WMMA_F32_16X16X4_F32.
// y[r] = sum_k W[r,k]*x[k] (+ bias[r]).  One wave = one 16-row tile.
// A layout (ISA 7.12.2, 32-bit A 16x4): lanes 0-15 hold K0/K1 in v[0]/v[1],
// lanes 16-31 hold K2/K3. B = vector broadcast across all 16 columns.
// EXEC is all-1s inside the wave (tile-uniform early exit; tail rows clamped).
// =====================================================================
__global__ void wmma_matvec(const float* __restrict__ W,
                            const float* __restrict__ x,
                            const float* __restrict__ bias,
                            float* __restrict__ y,
                            int R, int K) {
    const int lane = threadIdx.x & 31;
    const int wave = threadIdx.x >> 5;
    const int tile = blockIdx.x * (blockDim.x >> 5) + wave;
    const int row0 = tile * 16;
    if (row0 >= R) return;                 // wave-uniform: EXEC stays all-1s

    const int m  = lane & 15;
    const int hi = lane >> 4;              // 0: K0/K1,  1: K2/K3
    int row = row0 + m;
    if (row >= R) row = R - 1;             // tail tile: clamp (writes guarded)
    const float* wrow = W + (size_t)row * (size_t)K + (size_t)(hi * 2);
    const float* xk   = x + hi * 2;

    v8f acc = {};
    #pragma unroll 4
    for (int k = 0; k < K; k += 4) {
        v2f a, b;
        a.x = wrow[k];                     // 64-bit contiguous per lane
        a.y = wrow[k + 1];
        b.x = xk[k];                       // vector broadcast into B
        b.y = xk[k + 1];
        acc = __builtin_amdgcn_wmma_f32_16x16x4_f32(
            /*neg_a=*/false, a, /*neg_b=*/false, b,
            /*c_mod=*/(short)0, acc, /*reuse_a=*/false, /*reuse_b=*/false);
    }

    // D layout: lane n<16, VGPR r = D[M=r][N=n]; lanes 16-31 = rows 8..15.
    // B broadcast -> every column equal; lane 0 / lane 16 hold the results.
    if (m == 0) {
        const int base = row0 + hi * 8;
        #pragma unroll
        for (int r = 0; r < 8; ++r) {
            const int rr = base + r;
            if (rr < R) y[rr] = acc[r] + (bias ? bias[rr] : 0.0f);
        }
    }
}

// =====================================================================
// x = embedding[idx]
// =====================================================================
__global__ void gather_row(const float* __restrict__ emb,
                           const int* __restrict__ idx,
                           float* __restrict__ x) {
    int i = blockIdx.x * blockDim.x + threadIdx.x;
    long long row = (long long)idx[0];
    if (i < H) x[i] = emb[row * (long long)H + i];
}

// =====================================================================
// GRU gate fusion: h_new = (1-z)*n + z*h
// =====================================================================
__global__ void gru_gates(const float* __restrict__ gi,
                          const float* __restrict__ gh,
                          const float* __restrict__ h,
                          float* __restrict__ hnew,
                          float* __restrict__ hout) {
    int j = blockIdx.x * blockDim.x + threadIdx.x;
    if (j >= H) return;
    float r = 1.0f / (1.0f + expf(-(gi[j] + gh[j])));
    float z = 1.0f / (1.0f + expf(-(gi[H + j] + gh[H + j])));
    float n = tanhf(gi[2 * H + j] + r * gh[2 * H + j]);
    float hv = (1.0f - z) * n + z * h[j];
    hnew[j] = hv;
    hout[j] = hv;     // second tuple output: h_new
}

// =====================================================================
// softmax over 4096 scores, in place (single block, 1024 threads)
// =====================================================================
__global__ void attn_softmax(float* __restrict__ sc) {
    __shared__ float red[1024];
    const int tid = threadIdx.x;
    float m = -3.4e38f;
    for (int i = tid; i < L; i += 1024) m = fmaxf(m, sc[i]);
    red[tid] = m; __syncthreads();
    for (int s = 512; s > 0; s >>= 1) {
        if (tid < s) red[tid] = fmaxf(red[tid], red[tid + s]);
        __syncthreads();
    }
    const float gm = red[0]; __syncthreads();
    float sum = 0.0f;
    for (int i = tid; i < L; i += 1024) sum += expf(sc[i] - gm);
    red[tid] = sum; __syncthreads();
    for (int s = 512; s > 0; s >>= 1) {
        if (tid < s) red[tid] += red[tid + s];
        __syncthreads();
    }
    const float inv = 1.0f / red[0];
    for (int i = tid; i < L; i += 1024) sc[i] = expf(sc[i] - gm) * inv;
}

// =====================================================================
// ctx = attn @ enc : column reduction split into 16 L-chunks for parallelism
// =====================================================================
__global__ void ctx_partial(const float* __restrict__ attn,
                            const float* __restrict__ enc,
                            float* __restrict__ part) {
    const int k = blockIdx.x * blockDim.x + threadIdx.x;   // 0..1023 (coalesced)
    const int chunk = blockIdx.y;                          // 0..15
    const int l0 = chunk * (L / 16);
    float s = 0.0f;
    #pragma unroll 4
    for (int l = l0; l < l0 + L / 16; ++l)
        s += attn[l] * enc[(size_t)l * H + k];
    part[chunk * H + k] = s;
}

__global__ void ctx_reduce_concat(const float* __restrict__ part,
                                  const float* __restrict__ hn,
                                  float* __restrict__ cc) {
    const int k = blockIdx.x * blockDim.x + threadIdx.x;
    if (k >= H) return;
    float s = 0.0f;
    #pragma unroll
    for (int c = 0; c < 16; ++c) s += part[c * H + k];
    cc[k] = s;            // concat[0:H]   = ctx
    cc[H + k] = hn[k];    // concat[H:2H]  = h_new
}

// =====================================================================
// log-softmax over 50257 logits: 2-phase block reduction, then finalize
// =====================================================================
#define LS_CHUNK 786   // 64 blocks * 786 >= 50257
__global__ void ls_reduce1(const float* __restrict__ lg,
                           float* __restrict__ bmax, float* __restrict__ bsum,
                           int V) {
    __shared__ float red[256];
    const int b = blockIdx.x, tid = threadIdx.x;
    const int start = b * LS_CHUNK;
    const int end = min(V, start + LS_CHUNK);
    float vals[4]; float m = -3.4e38f;
    #pragma unroll
    for (int i = 0; i < 4; ++i) {
        const int idx = start + tid + i * 256;
        vals[i] = (idx < end) ? lg[idx] : -3.4e38f;
        m = fmaxf(m, vals[i]);
    }
    red[tid] = m; __syncthreads();
    for (int s = 128; s > 0; s >>= 1) {
        if (tid < s) red[tid] = fmaxf(red[tid], red[tid + s]);
        __syncthreads();
    }
    const float bm = red[0]; __syncthreads();
    float sum = 0.0f;
    #pragma unroll
    for (int i = 0; i < 4; ++i) {
        const int idx = start + tid + i * 256;
        if (idx < end) sum += expf(vals[i] - bm);
    }
    red[tid] = sum; __syncthreads();
    for (int s = 128; s > 0; s >>= 1) {
        if (tid < s) red[tid] += red[tid + s];
        __syncthreads();
    }
    if (tid == 0) { bmax[b] = bm; bsum[b] = red[0]; }
}

__global__ void ls_reduce2(const float* __restrict__ bmax,
                           const float* __restrict__ bsum,
                           float* __restrict__ g) {
    __shared__ float rm[64], rs[64];
    const int tid = threadIdx.x;     // 64 threads
    rm[tid] = bmax[tid]; __syncthreads();
    for (int s = 32; s > 0; s >>= 1) {
        if (tid < s) rm[tid] = fmaxf(rm[tid], rm[tid + s]);
        __syncthreads();
    }
    const float gm = rm[0];
    rs[tid] = bsum[tid] * expf(bmax[tid] - gm); __syncthreads();
    for (int s = 32; s > 0; s >>= 1) {
        if (tid < s) rs[tid] += rs[tid + s];
        __syncthreads();
    }
    if (tid == 0) { g[0] = gm; g[1] = logf(rs[0]); }
}

__global__ void ls_final(const float* __restrict__ lg,
                         const float* __restrict__ g,
                         float* __restrict__ out, int V) {
    const int v = blockIdx.x * blockDim.x + threadIdx.x;
    if (v < V) out[v] = lg[v] - g[0] - g[1];
}

// =====================================================================
extern "C" void kernel_launch(void* const* d_in, const int* in_sizes, int n_in,
                              void* d_out, int out_size, void* d_ws, size_t ws_size,
                              hipStream_t stream) {
    const int*   idx   = (const int*)  d_in[0];
    const float* hid   = (const float*)d_in[1];   // (1,1,H)
    const float* enc   = (const float*)d_in[2];   // (L,H)
    const float* emb   = (const float*)d_in[3];   // (V,H)
    const float* w_ih  = (const float*)d_in[4];   // (3H,H)
    const float* w_hh  = (const float*)d_in[5];   // (3H,H)
    const float* b_ih  = (const float*)d_in[6];
    const float* b_hh  = (const float*)d_in[7];
    const float* out_w = (const float*)d_in[8];   // (V,2H)
    const float* out_b = (const float*)d_in[9];
    float* out = (float*)d_out;                   // [V logsoftmax | H h_new]
    float* ws  = (float*)d_ws;

    float* X  = ws + WS_X;
    float* GI = ws + WS_GI;
    float* GH = ws + WS_GH;
    float* HN = ws + WS_HN;
    float* SC = ws + WS_SC;
    float* CP = ws + WS_CP;
    float* CC = ws + WS_CC;
    float* LG = ws + WS_LG;
    float* BM = ws + WS_BMAX;
    float* BS = ws + WS_BSUM;
    float* G  = ws + WS_G;

    // 1. embedding lookup
    gather_row<<<4, 256, 0, stream>>>(emb, idx, X);

    // 2. GRU pre-activations: gi = w_ih@x + b_ih ; gh = w_hh@h + b_hh
    //    3072 rows -> 192 tiles -> 24 blocks of 8 waves
    wmma_matvec<<<24, 256, 0, stream>>>(w_ih, X,   b_ih, GI, 3 * H, H);
    wmma_matvec<<<24, 256, 0, stream>>>(w_hh, hid, b_hh, GH, 3 * H, H);

    // 3. gates -> h_new (also second output)
    gru_gates<<<4, 256, 0, stream>>>(GI, GH, hid, HN, out + VOCAB);

    // 4. attention scores = enc @ h_new  (4096 rows -> 32 blocks)
    wmma_matvec<<<32, 256, 0, stream>>>(enc, HN, nullptr, SC, L, H);

    // 5. softmax(scores) in place
    attn_softmax<<<1, 1024, 0, stream>>>(SC);

    // 6. ctx = attn @ enc, then concat = [ctx, h_new]
    ctx_partial<<<dim3(4, 16), 256, 0, stream>>>(SC, enc, CP);
    ctx_reduce_concat<<<4, 256, 0, stream>>>(CP, HN, CC);

    // 7. logits = out_w @ concat + out_b   (412 MB stream — the HBM roofline)
    //    50257 rows -> 3142 tiles -> 393 blocks of 8 waves
    wmma_matvec<<<393, 256, 0, stream>>>(out_w, CC, out_b, LG, VOCAB, 2 * H);

    // 8. log_softmax(logits)
    ls_reduce1<<<64, 256, 0, stream>>>(LG, BM, BS, VOCAB);
    ls_reduce2<<<1, 64, 0, stream>>>(BM, BS, G);
    ls_final<<<(VOCAB + 255) / 256, 256, 0, stream>>>(LG, G, out, VOCAB);
}